// CubicBSpline_91053306675534
// MI455X (gfx1250) — compile-verified
//
#include <hip/hip_runtime.h>

// Tricubic B-spline evaluation on a 132^3 f32 grid, 2M points, 1 channel.
// Gather-bound: grid (9.2 MB) is L2/WGP$-resident on MI455X (192 MB L2),
// so we optimize VMEM issue count: one global_load_b128 per (i,j) row
// (z-stencil is contiguous), 16 loads per point, all in flight together.

#define GRIDN   132
#define MAXC    130.0f

typedef float v4f __attribute__((ext_vector_type(4)));

// 4B-aligned 16-byte load; gfx1250 handles sub-natural alignment for B128.
__device__ __forceinline__ v4f load_row4(const float* __restrict__ p) {
  v4f r;
  __builtin_memcpy(&r, p, sizeof(r));
  return r;
}

__device__ __forceinline__ void bspline_coefs(float x, float w[4]) {
  const float k = 1.0f / 6.0f;
  float omx = 1.0f - x;
  w[0] = omx * omx * omx * k;
  w[1] = fmaf(3.0f * (x - 2.0f), x * x, 4.0f) * k;
  w[2] = fmaf(-3.0f * omx * omx, x + 1.0f, 4.0f) * k;
  w[3] = x * x * x * k;
}

__global__ __launch_bounds__(256) void CubicBSpline_kernel(
    const float* __restrict__ pts,   // [n,3]
    const float* __restrict__ cp,    // [1,132,132,132]
    float* __restrict__ out,         // [n,1]
    int n)
{
  int p = blockIdx.x * blockDim.x + threadIdx.x;
  if (p >= n) return;

  // Streaming inputs: non-temporal so they don't evict the resident grid.
  float px = __builtin_nontemporal_load(pts + 3 * p + 0);
  float py = __builtin_nontemporal_load(pts + 3 * p + 1);
  float pz = __builtin_nontemporal_load(pts + 3 * p + 2);

  const float O = -(1.0f + 1e-08f);   // ORIGIN
  float tx = px - O;
  float ty = py - O;
  float tz = pz - O;

  bool inb = (tx >= 1.0f) & (ty >= 1.0f) & (tz >= 1.0f) &
             (tx < MAXC)  & (ty < MAXC)  & (tz < MAXC);
  if (!inb) { tx = 1.0f; ty = 1.0f; tz = 1.0f; }

  int ix = (int)tx, iy = (int)ty, iz = (int)tz;      // t>=1 -> trunc==floor
  float fx = tx - (float)ix;
  float fy = ty - (float)iy;
  float fz = tz - (float)iz;

  float wx[4], wy[4], wz[4];
  bspline_coefs(fx, wx);
  bspline_coefs(fy, wy);
  bspline_coefs(fz, wz);

  // Stencil base: (ix-1, iy-1, iz-1); always in [0,131] when clamped/in-bounds.
  const float* base = cp + ((size_t)(ix - 1) * GRIDN + (size_t)(iy - 1)) * GRIDN
                         + (size_t)(iz - 1);

  // Issue all 16 row gathers up front (one s_wait_loadcnt, max MLP).
  v4f rows[16];
#pragma unroll
  for (int i = 0; i < 4; ++i) {
#pragma unroll
    for (int j = 0; j < 4; ++j) {
      rows[i * 4 + j] = load_row4(base + ((size_t)i * GRIDN + (size_t)j) * GRIDN);
    }
  }

  // Factored contraction: z-dot, then y, then x.  ~84 FMAs/point.
  float acc = 0.0f;
#pragma unroll
  for (int i = 0; i < 4; ++i) {
    float si = 0.0f;
#pragma unroll
    for (int j = 0; j < 4; ++j) {
      v4f r = rows[i * 4 + j];
      float d = fmaf(r.x, wz[0],
                fmaf(r.y, wz[1],
                fmaf(r.z, wz[2], r.w * wz[3])));
      si = fmaf(wy[j], d, si);
    }
    acc = fmaf(wx[i], si, acc);
  }

  __builtin_nontemporal_store(inb ? acc : 0.0f, out + p);
}

extern "C" void kernel_launch(void* const* d_in, const int* in_sizes, int n_in,
                              void* d_out, int out_size, void* d_ws, size_t ws_size,
                              hipStream_t stream) {
  const float* pts = (const float*)d_in[0];   // 2M x 3 f32
  const float* cp  = (const float*)d_in[1];   // 1 x 132^3 f32
  float* out = (float*)d_out;                 // 2M x 1 f32

  int n = in_sizes[0] / 3;
  const int block = 256;                      // 8 wave32 waves / block
  int grid = (n + block - 1) / block;
  CubicBSpline_kernel<<<grid, block, 0, stream>>>(pts, cp, out, n);
}